// Net_27625229648073
// MI455X (gfx1250) — compile-verified
//
#include <hip/hip_runtime.h>
#include <float.h>
#include <math.h>

// ---------------------------------------------------------------------------
// GCN inference pipeline for gfx1250 (MI455X).
// Memory-bound (edge scatter ~2.2GB + ~82M f32 atomics @ 23.3TB/s HBM);
// dense transforms use the CDNA5 f32 WMMA path (V_WMMA_F32_16X16X4_F32) to
// keep full f32 fidelity. Post-pool layers exploit the fact that pooled
// features have only NG=8 distinct rows (8-row GEMM + table scatter).
// ---------------------------------------------------------------------------

typedef __attribute__((ext_vector_type(2))) float v2f;
typedef __attribute__((ext_vector_type(8))) float v8f;

#define BN_EPS 1e-3f

static __device__ __forceinline__ void atomicMaxF(float* addr, float val) {
  unsigned int* ua = (unsigned int*)addr;
  unsigned int old = *(volatile unsigned int*)ua;
  while (__uint_as_float(old) < val) {
    unsigned int assumed = old;
    old = atomicCAS(ua, assumed, __float_as_uint(val));
    if (old == assumed) break;
  }
}

// -------------------------------- utility ----------------------------------

__global__ void fill4_kernel(float4* __restrict__ p, float v, int n4) {
  int i = blockIdx.x * blockDim.x + threadIdx.x;
  if (i < n4) p[i] = float4{v, v, v, v};
}

__global__ void gsrc_kernel(const int* __restrict__ src, const int* __restrict__ ids,
                            int* __restrict__ gsrc, int nE) {
  int e = blockIdx.x * blockDim.x + threadIdx.x;
  if (e < nE) gsrc[e] = ids[src[e]];
}

// ------------------------- layer 1 transform (K=3) --------------------------

__global__ void gemm_in3(const float* __restrict__ x, const float* __restrict__ W1,
                         float* __restrict__ T, int nN) {
  int gid = blockIdx.x * blockDim.x + threadIdx.x;
  int n = gid >> 5;          // /32
  int f = gid & 31;          // %32
  if (n >= nN) return;
  const float* xr = x + n * 3;
  T[n * 32 + f] = xr[0] * W1[f] + xr[1] * W1[32 + f] + xr[2] * W1[64 + f];
}

// --------------------- WMMA f32 GEMM: T[N,FO] = X[N,FI]@W ------------------
// One wave (32 threads) per 16x16 output tile; K-loop in steps of 4 using
// V_WMMA_F32_16X16X4_F32.  VGPR layouts per CDNA5 ISA 7.12.2:
//   A (16x4):  lane L -> m=L%16, hi=L/16 ; vgpr r holds A[m][k + r + 2*hi]
//   B (4x16):  lane L -> n=L%16, hi=L/16 ; vgpr r holds B[k + r + 2*hi][n]
//   C/D:       lane L -> n=L%16, hi=L/16 ; vgpr r holds C[r + 8*hi][n]

template <int FI, int FO>
__global__ void gemm_wmma(const float* __restrict__ X, const float* __restrict__ W,
                          float* __restrict__ T, int nN) {
  const int lane = threadIdx.x;       // 0..31, one wave per block
  const int m    = lane & 15;
  const int hi   = lane >> 4;         // 0 or 1
  const int row0 = blockIdx.x * 16;
  const int col0 = blockIdx.y * 16;

  int arow = row0 + m;
  if (arow >= nN) arow = nN - 1;      // clamp loads; stores guarded below
  const float* xr = X + (size_t)arow * FI + 2 * hi;

  v8f acc = {};
#pragma unroll
  for (int k = 0; k < FI; k += 4) {
    v2f a = *(const v2f*)(xr + k);    // 8B-aligned: two contiguous K values
    v2f b;
    b.x = W[(k + 2 * hi + 0) * FO + col0 + m];
    b.y = W[(k + 2 * hi + 1) * FO + col0 + m];
    acc = __builtin_amdgcn_wmma_f32_16x16x4_f32(false, a, false, b,
                                                (short)0, acc, false, false);
  }

  if (row0 + 16 <= nN) {
    // wave-uniform fast path: 8 stores with immediate offsets off one base
    float* tb = T + (size_t)(row0 + 8 * hi) * FO + col0 + m;
#pragma unroll
    for (int r = 0; r < 8; ++r) tb[r * FO] = acc[r];
  } else {
#pragma unroll
    for (int r = 0; r < 8; ++r) {
      int row = row0 + r + 8 * hi;
      if (row < nN) T[(size_t)row * FO + col0 + m] = acc[r];
    }
  }
}

// ------------------------------ edge scatter --------------------------------
// One thread per (edge, 4-feature group): b128 feature load, one set of
// index/weight loads per 4 features, 4 consecutive-address f32 atomics
// (mergeable per cacheline at L2).

template <int FO>
__global__ void scatter_edges4(const float* __restrict__ T, const int* __restrict__ src,
                               const int* __restrict__ dst, const float* __restrict__ ew,
                               float* __restrict__ agg, int nE) {
  constexpr int G = FO / 4;
  int gid = blockIdx.x * blockDim.x + threadIdx.x;
  int e = gid / G;
  int g = gid % G;
  if (e >= nE) return;
  float w = ew[e];
  float4 t = *(const float4*)(T + (size_t)src[e] * FO + g * 4);
  float* ap = agg + (size_t)dst[e] * FO + g * 4;
  atomicAdd(ap + 0, t.x * w);
  atomicAdd(ap + 1, t.y * w);
  atomicAdd(ap + 2, t.z * w);
  atomicAdd(ap + 3, t.w * w);
}

// scatter where transformed features have only 8 distinct rows (post-pool)
template <int FO>
__global__ void scatter_pooled4(const float* __restrict__ Q8, const int* __restrict__ gsrc,
                                const int* __restrict__ dst, const float* __restrict__ ew,
                                float* __restrict__ agg, int nE) {
  constexpr int G = FO / 4;
  int gid = blockIdx.x * blockDim.x + threadIdx.x;
  int e = gid / G;
  int g = gid % G;
  if (e >= nE) return;
  float w = ew[e];
  float4 t = *(const float4*)(Q8 + gsrc[e] * FO + g * 4);
  float* ap = agg + (size_t)dst[e] * FO + g * 4;
  atomicAdd(ap + 0, t.x * w);
  atomicAdd(ap + 1, t.y * w);
  atomicAdd(ap + 2, t.z * w);
  atomicAdd(ap + 3, t.w * w);
}

// --------------------------- elementwise epilogues --------------------------

template <int F>
__global__ void bias_relu4(float* __restrict__ h, const float* __restrict__ b, int n4) {
  int i = blockIdx.x * blockDim.x + threadIdx.x;
  if (i >= n4) return;
  int f4 = i & (F / 4 - 1);
  float4 bb = ((const float4*)b)[f4];
  float4 x = ((float4*)h)[i];
  x.x = fmaxf(x.x + bb.x, 0.0f);
  x.y = fmaxf(x.y + bb.y, 0.0f);
  x.z = fmaxf(x.z + bb.z, 0.0f);
  x.w = fmaxf(x.w + bb.w, 0.0f);
  ((float4*)h)[i] = x;
}

template <int F>
__global__ void bias_relu_bn4(float* __restrict__ h, const float* __restrict__ b,
                              const float* __restrict__ g, const float* __restrict__ be,
                              const float* __restrict__ mu, const float* __restrict__ va,
                              int n4) {
  int i = blockIdx.x * blockDim.x + threadIdx.x;
  if (i >= n4) return;
  int f4 = i & (F / 4 - 1);
  float4 bb = ((const float4*)b)[f4];
  float4 gg = ((const float4*)g)[f4];
  float4 ee = ((const float4*)be)[f4];
  float4 mm = ((const float4*)mu)[f4];
  float4 vv = ((const float4*)va)[f4];
  float4 x = ((float4*)h)[i];
  x.x = (fmaxf(x.x + bb.x, 0.0f) - mm.x) * (gg.x * rsqrtf(vv.x + BN_EPS)) + ee.x;
  x.y = (fmaxf(x.y + bb.y, 0.0f) - mm.y) * (gg.y * rsqrtf(vv.y + BN_EPS)) + ee.y;
  x.z = (fmaxf(x.z + bb.z, 0.0f) - mm.z) * (gg.z * rsqrtf(vv.z + BN_EPS)) + ee.z;
  x.w = (fmaxf(x.w + bb.w, 0.0f) - mm.w) * (gg.w * rsqrtf(vv.w + BN_EPS)) + ee.w;
  ((float4*)h)[i] = x;
}

// ------------------------------ segment max pool ----------------------------
// ids sorted: each block walks a chunk of nodes with coalesced feature reads,
// flushing a float-atomicMax only at graph boundaries (few hundred atomics).

template <int F, int CHUNK>
__global__ void pool_max(const float* __restrict__ H, const int* __restrict__ ids,
                         float* __restrict__ P, int nN) {
  const int f  = threadIdx.x;     // blockDim.x == F
  const int n0 = blockIdx.x * CHUNK;
  int n1 = n0 + CHUNK;
  if (n1 > nN) n1 = nN;
  int curg = ids[n0];
  float mx = -FLT_MAX;
  for (int n = n0; n < n1; ++n) {
    int gg = ids[n];
    if (gg != curg) {
      atomicMaxF(&P[curg * F + f], mx);
      curg = gg;
      mx = -FLT_MAX;
    }
    mx = fmaxf(mx, H[(size_t)n * F + f]);
  }
  atomicMaxF(&P[curg * F + f], mx);
}

// -------------------- tiny 8-row GEMM for pooled transforms -----------------

template <int FI, int FO>
__global__ void gemm8(const float* __restrict__ P, const float* __restrict__ W,
                      float* __restrict__ Q) {
  int idx = threadIdx.x;          // 8*FO threads, one block
  if (idx >= 8 * FO) return;
  int g = idx / FO, f = idx % FO;
  float s = 0.0f;
#pragma unroll 8
  for (int k = 0; k < FI; ++k) s += P[g * FI + k] * W[k * FO + f];
  Q[idx] = s;
}

// ------------------------------ bias + softmax ------------------------------

__global__ void softmax20(const float* __restrict__ A, const float* __restrict__ b5,
                          float* __restrict__ out, int nN) {
  int n = blockIdx.x * blockDim.x + threadIdx.x;
  if (n >= nN) return;
  float v[20];
  const float4* a4 = reinterpret_cast<const float4*>(A + (size_t)n * 20);
#pragma unroll
  for (int j = 0; j < 5; ++j) {
    float4 t = a4[j];
    v[4 * j + 0] = t.x; v[4 * j + 1] = t.y; v[4 * j + 2] = t.z; v[4 * j + 3] = t.w;
  }
  float mx = -FLT_MAX;
#pragma unroll
  for (int j = 0; j < 20; ++j) { v[j] += b5[j]; mx = fmaxf(mx, v[j]); }
  float s = 0.0f;
#pragma unroll
  for (int j = 0; j < 20; ++j) { v[j] = expf(v[j] - mx); s += v[j]; }
  float inv = 1.0f / s;
  float4* o4 = reinterpret_cast<float4*>(out + (size_t)n * 20);
#pragma unroll
  for (int j = 0; j < 5; ++j) {
    float4 t;
    t.x = v[4 * j + 0] * inv; t.y = v[4 * j + 1] * inv;
    t.z = v[4 * j + 2] * inv; t.w = v[4 * j + 3] * inv;
    o4[j] = t;
  }
}

// ---------------------------------------------------------------------------

static inline int ceil_div(long long a, long long b) { return (int)((a + b - 1) / b); }

extern "C" void kernel_launch(void* const* d_in, const int* in_sizes, int n_in,
                              void* d_out, int out_size, void* d_ws, size_t ws_size,
                              hipStream_t stream) {
  const float* x   = (const float*)d_in[0];
  const float* ew  = (const float*)d_in[1];
  const int*   src = (const int*)d_in[2];
  const int*   dst = (const int*)d_in[3];
  const int*   ids = (const int*)d_in[4];
  const float* W1 = (const float*)d_in[5],  *b1 = (const float*)d_in[6];
  const float* W2 = (const float*)d_in[7],  *b2 = (const float*)d_in[8];
  const float* g1 = (const float*)d_in[9],  *be1 = (const float*)d_in[10];
  const float* m1 = (const float*)d_in[11], *v1 = (const float*)d_in[12];
  const float* W3 = (const float*)d_in[13], *b3 = (const float*)d_in[14];
  const float* W4 = (const float*)d_in[15], *b4 = (const float*)d_in[16];
  const float* g2 = (const float*)d_in[17], *be2 = (const float*)d_in[18];
  const float* m2 = (const float*)d_in[19], *v2 = (const float*)d_in[20];
  const float* W5 = (const float*)d_in[21], *b5 = (const float*)d_in[22];

  const int N = in_sizes[0] / 3;   // 80000 (multiple of 16)
  const int E = in_sizes[1];       // 1280000
  float* out = (float*)d_out;

  // workspace layout
  float* A    = (float*)d_ws;                 // N*64 floats
  float* B    = A + (size_t)N * 64;           // N*64 floats
  int*   gsrc = (int*)(B + (size_t)N * 64);   // E ints
  float* P    = (float*)(gsrc + E);           // 8*64 floats (pool maxima)
  float* Q    = P + 8 * 64;                   // 8*64 floats (pooled transform)

  const int BS = 256;

  // graph id of each edge's source node (constant; used by layers 3 & 5)
  gsrc_kernel<<<ceil_div(E, BS), BS, 0, stream>>>(src, ids, gsrc, E);

  // ---- Layer 1: relu(Agg(x@W1) + b1), F=32
  gemm_in3<<<ceil_div((long long)N * 32, BS), BS, 0, stream>>>(x, W1, A, N);
  fill4_kernel<<<ceil_div((long long)N * 8, BS), BS, 0, stream>>>((float4*)B, 0.0f, N * 8);
  scatter_edges4<32><<<ceil_div((long long)E * 8, BS), BS, 0, stream>>>(A, src, dst, ew, B, E);
  bias_relu4<32><<<ceil_div((long long)N * 8, BS), BS, 0, stream>>>(B, b1, N * 8);

  // ---- Layer 2: bn1(relu(Agg(h@W2) + b2)), F=32  (WMMA GEMM)
  {
    dim3 grid(ceil_div(N, 16), 32 / 16);
    gemm_wmma<32, 32><<<grid, 32, 0, stream>>>(B, W2, A, N);
  }
  fill4_kernel<<<ceil_div((long long)N * 8, BS), BS, 0, stream>>>((float4*)B, 0.0f, N * 8);
  scatter_edges4<32><<<ceil_div((long long)E * 8, BS), BS, 0, stream>>>(A, src, dst, ew, B, E);
  bias_relu_bn4<32><<<ceil_div((long long)N * 8, BS), BS, 0, stream>>>(B, b2, g1, be1, m1, v1, N * 8);

  // ---- Pool 1 (8 graphs x 32)
  fill4_kernel<<<1, 64, 0, stream>>>((float4*)P, -FLT_MAX, 8 * 8);
  pool_max<32, 256><<<ceil_div(N, 256), 32, 0, stream>>>(B, ids, P, N);

  // ---- Layer 3: input has 8 distinct rows -> 8-row GEMM + table scatter
  gemm8<32, 64><<<1, 512, 0, stream>>>(P, W3, Q);
  fill4_kernel<<<ceil_div((long long)N * 16, BS), BS, 0, stream>>>((float4*)B, 0.0f, N * 16);
  scatter_pooled4<64><<<ceil_div((long long)E * 16, BS), BS, 0, stream>>>(Q, gsrc, dst, ew, B, E);
  bias_relu4<64><<<ceil_div((long long)N * 16, BS), BS, 0, stream>>>(B, b3, N * 16);

  // ---- Layer 4: bn2(relu(Agg(h@W4) + b4)), F=64  (WMMA GEMM)
  {
    dim3 grid(ceil_div(N, 16), 64 / 16);
    gemm_wmma<64, 64><<<grid, 32, 0, stream>>>(B, W4, A, N);
  }
  fill4_kernel<<<ceil_div((long long)N * 16, BS), BS, 0, stream>>>((float4*)B, 0.0f, N * 16);
  scatter_edges4<64><<<ceil_div((long long)E * 16, BS), BS, 0, stream>>>(A, src, dst, ew, B, E);
  bias_relu_bn4<64><<<ceil_div((long long)N * 16, BS), BS, 0, stream>>>(B, b4, g2, be2, m2, v2, N * 16);

  // ---- Pool 2 (8 graphs x 64)
  fill4_kernel<<<1, 128, 0, stream>>>((float4*)P, -FLT_MAX, 8 * 16);
  pool_max<64, 256><<<ceil_div(N, 256), 64, 0, stream>>>(B, ids, P, N);

  // ---- Layer 5: 8-row GEMM [8,64]@[64,20] + scatter + bias + softmax
  gemm8<64, 20><<<1, 160, 0, stream>>>(P, W5, Q);
  fill4_kernel<<<ceil_div((long long)N * 5, BS), BS, 0, stream>>>((float4*)A, 0.0f, N * 5);
  scatter_pooled4<20><<<ceil_div((long long)E * 5, BS), BS, 0, stream>>>(Q, gsrc, dst, ew, A, E);
  softmax20<<<ceil_div(N, BS), BS, 0, stream>>>(A, b5, out, N);
}